// NTKernel_2276332667480
// MI455X (gfx1250) — compile-verified
//
#include <hip/hip_runtime.h>
#include <hip/hip_bf16.h>
#include <math.h>

#define BATCH 512
#define D_IN  128
#define D_H   256
#define D_OUTC 10   // W3 is [256,10], we differentiate channel 0

typedef __attribute__((ext_vector_type(2))) float v2f;
typedef __attribute__((ext_vector_type(8))) float v8f;

// ---------------------------------------------------------------------------
// Kernel 1: per-sample forward + backward. One block (256 thr) per sample.
// Produces h1, h2, delta1, delta2 ([512,256] each) and per-sample Jacobian
// row norm:  n^2 = (x.x+1)*d1.d1 + (h1.h1+1)*d2.d2 + h2.h2 + 1
// ---------------------------------------------------------------------------
__global__ __launch_bounds__(256) void ntk_prep(
    const float* __restrict__ X,
    const float* __restrict__ W1, const float* __restrict__ b1,
    const float* __restrict__ W2, const float* __restrict__ b2,
    const float* __restrict__ W3,
    float* __restrict__ h1o, float* __restrict__ h2o,
    float* __restrict__ d1o, float* __restrict__ d2o,
    float* __restrict__ normo)
{
    __shared__ float x_s[D_IN];
    __shared__ float h1_s[D_H];
    __shared__ float d2_s[D_H];
    __shared__ float acc[5];   // xx, h1h1, h2h2, d1d1, d2d2

    const int j = threadIdx.x;   // hidden unit handled by this thread
    const int s = blockIdx.x;    // sample

    if (j < D_IN) x_s[j] = X[s * D_IN + j];
    if (j < 5)    acc[j] = 0.0f;
    __syncthreads();

    // layer 1: z1[j] = b1[j] + sum_i x[i] * W1[i, j]   (coalesced over j)
    float z1 = b1[j];
    #pragma unroll 8
    for (int i = 0; i < D_IN; ++i) z1 = fmaf(x_s[i], W1[i * D_H + j], z1);
    const float h1 = z1 > 0.0f ? z1 : 0.0f;
    h1_s[j] = h1;
    __syncthreads();

    // layer 2: z2[j] = b2[j] + sum_i h1[i] * W2[i, j]
    float z2 = b2[j];
    #pragma unroll 8
    for (int i = 0; i < D_H; ++i) z2 = fmaf(h1_s[i], W2[i * D_H + j], z2);
    const float h2  = z2 > 0.0f ? z2 : 0.0f;
    const float dd2 = (z2 > 0.0f) ? W3[j * D_OUTC + 0] : 0.0f;   // delta2
    d2_s[j] = dd2;
    __syncthreads();

    // delta1[j] = 1(z1>0) * sum_k W2[j, k] * delta2[k]
    float dd1 = 0.0f;
    if (z1 > 0.0f) {
        #pragma unroll 8
        for (int k = 0; k < D_H; ++k) dd1 = fmaf(W2[j * D_H + k], d2_s[k], dd1);
    }

    h1o[s * D_H + j] = h1;
    h2o[s * D_H + j] = h2;
    d1o[s * D_H + j] = dd1;
    d2o[s * D_H + j] = dd2;

    if (j < D_IN) atomicAdd(&acc[0], x_s[j] * x_s[j]);
    atomicAdd(&acc[1], h1 * h1);
    atomicAdd(&acc[2], h2 * h2);
    atomicAdd(&acc[3], dd1 * dd1);
    atomicAdd(&acc[4], dd2 * dd2);
    __syncthreads();

    if (j == 0) {
        float nsq = (acc[0] + 1.0f) * acc[3] + (acc[1] + 1.0f) * acc[4]
                  + acc[2] + 1.0f;
        normo[s] = sqrtf(nsq);
    }
}

// ---------------------------------------------------------------------------
// Kernel 2: fused 5x Gram + NTK combine + normalization.
// One wave32 per 16x16 output tile; fp32 WMMA 16x16x4.
// For G = F1 @ F2^T both operands use the identical per-lane load pattern:
//   lane L reads float2 at F[(tile*16 + (L&15))*K + k + 2*(L>>4)]
// D layout: lane L holds column (L&15), rows r + 8*(L>>4), r = 0..7.
// ---------------------------------------------------------------------------
__device__ __forceinline__ v8f wmma_f32(v2f a, v2f b, v8f c) {
    return __builtin_amdgcn_wmma_f32_16x16x4_f32(
        /*neg_a=*/false, a, /*neg_b=*/false, b,
        /*c_mod=*/(short)0, c, /*reuse_a=*/false, /*reuse_b=*/false);
}

__global__ __launch_bounds__(32) void ntk_gram(
    const float* __restrict__ x1,  const float* __restrict__ x2,
    const float* __restrict__ h1a, const float* __restrict__ h2a,
    const float* __restrict__ d1a, const float* __restrict__ d2a,
    const float* __restrict__ h1b, const float* __restrict__ h2b,
    const float* __restrict__ d1b, const float* __restrict__ d2b,
    const float* __restrict__ n1v, const float* __restrict__ n2v,
    float* __restrict__ out)
{
    const int lane = threadIdx.x & 31;
    const int l16  = lane & 15;
    const int half = lane >> 4;          // 0 or 1
    const int koff = half * 2;

    const int mrow = blockIdx.x * 16 + l16;   // row index into side-1 features
    const int nrow = blockIdx.y * 16 + l16;   // row index into side-2 features

    v8f gx  = {};
    v8f gh1 = {};
    v8f gh2 = {};
    v8f gd1 = {};
    v8f gd2 = {};

    // x-Gram over K = 128
    const float* __restrict__ pa = x1 + mrow * D_IN + koff;
    const float* __restrict__ pb = x2 + nrow * D_IN + koff;
    #pragma unroll 4
    for (int k = 0; k < D_IN; k += 4) {
        v2f a = *(const v2f*)(pa + k);
        v2f b = *(const v2f*)(pb + k);
        gx = wmma_f32(a, b, gx);
    }

    // four feature Grams over K = 256
    const int mo = mrow * D_H + koff;
    const int no = nrow * D_H + koff;
    #pragma unroll 2
    for (int k = 0; k < D_H; k += 4) {
        v2f a, b;
        a = *(const v2f*)(h1a + mo + k); b = *(const v2f*)(h1b + no + k);
        gh1 = wmma_f32(a, b, gh1);
        a = *(const v2f*)(h2a + mo + k); b = *(const v2f*)(h2b + no + k);
        gh2 = wmma_f32(a, b, gh2);
        a = *(const v2f*)(d1a + mo + k); b = *(const v2f*)(d1b + no + k);
        gd1 = wmma_f32(a, b, gd1);
        a = *(const v2f*)(d2a + mo + k); b = *(const v2f*)(d2b + no + k);
        gd2 = wmma_f32(a, b, gd2);
    }

    // combine:  K = (Gx+1)*Gd1 + (Gh1+1)*Gd2 + Gh2 + 1, then row/col normalize
    const int on    = blockIdx.y * 16 + l16;
    const int mbase = blockIdx.x * 16 + half * 8;
    const float rn2 = n2v[on];
    #pragma unroll
    for (int r = 0; r < 8; ++r) {
        const int om = mbase + r;
        float v = (gx[r] + 1.0f) * gd1[r] + (gh1[r] + 1.0f) * gd2[r]
                + gh2[r] + 1.0f;
        out[om * BATCH + on] = v / (n1v[om] * rn2);
    }
}

// ---------------------------------------------------------------------------
extern "C" void kernel_launch(void* const* d_in, const int* in_sizes, int n_in,
                              void* d_out, int out_size, void* d_ws, size_t ws_size,
                              hipStream_t stream) {
    const float* x1 = (const float*)d_in[0];
    const float* x2 = (const float*)d_in[1];
    const float* W1 = (const float*)d_in[2];
    const float* b1 = (const float*)d_in[3];
    const float* W2 = (const float*)d_in[4];
    const float* b2 = (const float*)d_in[5];
    const float* W3 = (const float*)d_in[6];
    // d_in[7] = b3: gradients do not depend on it (db3 = e_C, constant)

    const size_t FEAT = (size_t)BATCH * D_H;   // 131072 floats
    float* ws  = (float*)d_ws;
    float* h1a = ws;
    float* h2a = h1a + FEAT;
    float* d1a = h2a + FEAT;
    float* d2a = d1a + FEAT;
    float* h1b = d2a + FEAT;
    float* h2b = h1b + FEAT;
    float* d1b = h2b + FEAT;
    float* d2b = d1b + FEAT;
    float* n1v = d2b + FEAT;          // 512
    float* n2v = n1v + BATCH;         // 512   (total ~4.2 MB)

    ntk_prep<<<BATCH, 256, 0, stream>>>(x1, W1, b1, W2, b2, W3,
                                        h1a, h2a, d1a, d2a, n1v);
    ntk_prep<<<BATCH, 256, 0, stream>>>(x2, W1, b1, W2, b2, W3,
                                        h1b, h2b, d1b, d2b, n2v);

    dim3 grid(BATCH / 16, BATCH / 16);   // 32 x 32 tiles
    ntk_gram<<<grid, 32, 0, stream>>>(x1, x2,
                                      h1a, h2a, d1a, d2a,
                                      h1b, h2b, d1b, d2b,
                                      n1v, n2v, (float*)d_out);
}